// StateSpaceSimulatorAugmented_MANAM_120259084399
// MI455X (gfx1250) — compile-verified
//
#include <hip/hip_runtime.h>
#include <math.h>

// ---------------------------------------------------------------------------
// State-space scan, B=512 x T=2048, NXF=NXB=16, NU=8, NY=8.
// One wave32 per 16-row batch tile; f32 WMMA 16x16x4 for all mat-vecs.
// Weights live in VGPR B-fragments; state relayout via one LDS tile/step;
// all t-dependent scalars come from a precomputed LDS table.
// ---------------------------------------------------------------------------

typedef float v2f __attribute__((ext_vector_type(2)));
typedef float v8f __attribute__((ext_vector_type(8)));

#define B_   512
#define T_   2048
#define LSTR 20   // LDS tile row stride (floats): conflict-free for our patterns
#define TSTR 6    // scalar-table stride (floats): 3x 8-byte-aligned pairs

static __device__ __forceinline__ v8f wmma4(v2f a, v2f b, v8f c) {
  // D(16x16,f32) = A(16x4,f32) * B(4x16,f32) + C
  return __builtin_amdgcn_wmma_f32_16x16x4_f32(false, a, false, b,
                                               (short)0, c, false, false);
}

#if __has_builtin(__builtin_amdgcn_tanhf)
static __device__ __forceinline__ float fast_tanh(float x) {
  return __builtin_amdgcn_tanhf(x);   // V_TANH_F32 (gfx1250 transcendental)
}
#else
static __device__ __forceinline__ float fast_tanh(float x) { return tanhf(x); }
#endif

static __device__ __forceinline__ float clampf(float v, float lo, float hi) {
  return fminf(fmaxf(v, lo), hi);
}

__global__ __launch_bounds__(32, 1) void ssm_scan_kernel(
    const float* __restrict__ xf0p, const float* __restrict__ xb0p,
    const float* __restrict__ up,
    const float* __restrict__ theta_f, const float* __restrict__ theta_h,
    const float* __restrict__ datascaler,
    const float* __restrict__ Wf, const float* __restrict__ bf,
    const float* __restrict__ Wfb, const float* __restrict__ Wgb,
    const float* __restrict__ Wh, const float* __restrict__ Whb,
    const float* __restrict__ min_xb, const float* __restrict__ max_xb,
    const float* __restrict__ min_x, const float* __restrict__ max_x,
    const float* __restrict__ min_y, const float* __restrict__ max_y,
    float* __restrict__ out)
{
  __shared__ __align__(16) float buf[2][16 * LSTR]; // xf / xb staging tiles
  __shared__ __align__(16) float tbl[T_ * TSTR];    // per-t scalars (~48 KB)

  const int lane = threadIdx.x;   // 0..31
  const int n = lane & 15;        // column of C/D tile owned by this lane
  const int h = lane >> 4;        // half-wave select
  const int m = n;                // A-matrix row (batch row within tile)
  const int b0 = blockIdx.x << 4; // first batch row of this tile

  // ---- per-t scalar table: {inv2, r21, c1, c2, d4c, tf2} ----
  for (int t = lane; t < T_; t += 32) {
    float tf = (float)t;
    float tf1 = tf + 1.0f, tf2 = tf + 2.0f;
    float* p = &tbl[t * TSTR];
    p[0] = 1.0f / tf2;                    // ref: 1/(t+2)
    p[1] = tf2 / tf1;                     // (t+2)/(t+1)
    p[2] = tf1 / tf2;                     // (t+1)/(t+2)
    p[3] = tf1 / (tf2 * tf2);             // (t+1)/(t+2)^2
    p[4] = 2.0f * tf2 * tf2 / tf1;        // 2(t+2)^2/(t+1)
    p[5] = tf2;
  }

  const float ds = datascaler[0];
  const float bft = bf[n] + theta_f[n];            // bias of f(), per column
  const float th  = (n < 8) ? theta_h[n] : 0.0f;   // bias of h(), per column
  const float cminxb = min_xb[0], cmaxxb = max_xb[0];
  const float cminx  = min_x[0],  cmaxx  = max_x[0];
  const float cminy  = min_y[0],  cmaxy  = max_y[0];

  // ---- build weight B-fragments once; they stay in VGPRs across the scan ----
  // B(4x16) fragment layout (mirrors A): lane holds N=lane%16,
  // VGPR0 <-> K=k0+2h, VGPR1 <-> K=k0+2h+1.
  v2f wf_f[6], wfb_f[10], wgb_f[10], wc_f[8];
#pragma unroll
  for (int i = 0; i < 6; ++i) {          // Wf' (24x16): rows<16 scaled by ds
    int k = 4 * i + 2 * h;
    float s0 = (k     < 16) ? ds : 1.0f;
    float s1 = (k + 1 < 16) ? ds : 1.0f;
    wf_f[i].x = s0 * Wf[(k    ) * 16 + n];
    wf_f[i].y = s1 * Wf[(k + 1) * 16 + n];
  }
#pragma unroll
  for (int i = 0; i < 10; ++i) {         // Wfb, Wgb (40x16)
    int k = 4 * i + 2 * h;
    wfb_f[i].x = Wfb[(k    ) * 16 + n];
    wfb_f[i].y = Wfb[(k + 1) * 16 + n];
    wgb_f[i].x = Wgb[(k    ) * 16 + n];
    wgb_f[i].y = Wgb[(k + 1) * 16 + n];
  }
#pragma unroll
  for (int i = 0; i < 8; ++i) {          // Wcomb (32x16): ds*Wh + Whb, padded N
    int k = 4 * i + 2 * h;               // 0..31
    float v0 = 0.0f, v1 = 0.0f;
    if (n < 8) {
      v0 = Whb[(k    ) * 8 + n] + ((k     < 16) ? ds * Wh[(k    ) * 8 + n] : 0.0f);
      v1 = Whb[(k + 1) * 8 + n] + ((k + 1 < 16) ? ds * Wh[(k + 1) * 8 + n] : 0.0f);
    }
    wc_f[i].x = v0; wc_f[i].y = v1;
  }

  // ---- initial state in C/D tile layout: lane holds col n, rows r+8h ----
  v8f xf, xb, mu, sg, y;
#pragma unroll
  for (int r = 0; r < 8; ++r) {
    int row = b0 + r + 8 * h;
    xf[r] = xf0p[row * 16 + n];
    xb[r] = xb0p[row * 16 + n];
  }
  mu = xb;
#pragma unroll
  for (int r = 0; r < 8; ++r) sg[r] = 1.0f;

  // stage initial carry; its fragments serve both y0 and iteration 0
#pragma unroll
  for (int r = 0; r < 8; ++r) {
    int row = r + 8 * h;
    buf[0][row * LSTR + n] = xf[r];
    buf[1][row * LSTR + n] = xb[r];
  }
  __syncthreads(); // single-wave WG: barrier is a NOP, orders LDS ops

  v2f zx[4], zb[4];           // A-fragments of current carry (xf | xb)
#pragma unroll
  for (int i = 0; i < 4; ++i) {
    zx[i] = *reinterpret_cast<const v2f*>(&buf[0][m * LSTR + 4 * i + 2 * h]);
    zb[i] = *reinterpret_cast<const v2f*>(&buf[1][m * LSTR + 4 * i + 2 * h]);
  }

  // ---- y0 = h(xf0) + h_b(xf0, xb0) (unclamped, per reference carry0) ----
  {
    v8f a0 = {}, a1 = {};
#pragma unroll
    for (int i = 0; i < 4; ++i) {
      a0 = wmma4(zx[i], wc_f[i],     a0);
      a1 = wmma4(zb[i], wc_f[4 + i], a1);
    }
#pragma unroll
    for (int r = 0; r < 8; ++r) y[r] = a0[r] + a1[r] + th;
  }

  // ---- streaming pointers (strength-reduced; row offsets go in IOFFSET) ----
  const long long OFF1 = (long long)B_ * T_ * 16;
  const long long OFF2 = 2 * OFF1;
  float* pxf = out +        ((long long)(b0 + 8 * h) * T_) * 16 + n;
  float* pxb = out + OFF1 + ((long long)(b0 + 8 * h) * T_) * 16 + n;
  float* py  = out + OFF2 + ((long long)(b0 + 8 * h) * T_) * 8  + n;
  const float* pu = up + ((long long)(b0 + m) * T_) * 8 + 2 * h;

#pragma unroll 1
  for (int t = 0; t < T_; ++t) {
    // per-step lane-uniform scalars from the LDS table (broadcast reads)
    v2f sA = *reinterpret_cast<const v2f*>(&tbl[t * TSTR]);     // inv2, r21
    v2f sB = *reinterpret_cast<const v2f*>(&tbl[t * TSTR + 2]); // c1, c2
    v2f sC = *reinterpret_cast<const v2f*>(&tbl[t * TSTR + 4]); // d4c, tf2
    const float inv2 = sA.x, r21 = sA.y;
    const float c1 = sB.x, c2 = sB.y;
    const float d4c = sC.x, tf2 = sC.y;

    // ---- emit carry (states BEFORE update), streaming/non-temporal ----
#pragma unroll
    for (int r = 0; r < 8; ++r) {
      __builtin_nontemporal_store(xf[r], pxf + r * (T_ * 16));
      __builtin_nontemporal_store(xb[r], pxb + r * (T_ * 16));
    }
    if (n < 8) {
#pragma unroll
      for (int r = 0; r < 8; ++r)
        __builtin_nontemporal_store(y[r], py + r * (T_ * 8));
    }

    // ---- u A-fragments + prefetch of the u stream ----
    v2f z8 = *reinterpret_cast<const v2f*>(pu);       // u K 0..3
    v2f z9 = *reinterpret_cast<const v2f*>(pu + 4);   // u K 4..7
    __builtin_prefetch(pu + 8 * 32, 0, 1);            // ~32 steps ahead

    // ---- carry-only element terms: compute BEFORE the WMMA chains so the
    //      1/sigma division and eps base overlap with matrix work ----
    v8f igpv, devv, e1v;
#pragma unroll
    for (int r = 0; r < 8; ++r) {
      igpv[r] = 1.0f / sg[r];                 // ref: 1/gamma_plus (exact div)
      devv[r] = xb[r] - mu[r];
      e1v[r]  = tf2 * (sg[r] * sg[r] - r21);  // (t+2)*(sigma^2-(t+2)/(t+1))
    }

    // ---- pre-activations: f (K=24), f_b, g_b (K=40); split accumulator
    //      chains (depth 5 instead of 10) for WMMA ILP ----
    v8f f0 = {}, f1 = {}, fb0 = {}, fb1 = {}, gb0 = {}, gb1 = {};
#pragma unroll
    for (int i = 0; i < 4; ++i) {
      fb0 = wmma4(zx[i], wfb_f[i],     fb0);
      gb0 = wmma4(zx[i], wgb_f[i],     gb0);
      fb1 = wmma4(zb[i], wfb_f[4 + i], fb1);
      gb1 = wmma4(zb[i], wgb_f[4 + i], gb1);
    }
    f0 = wmma4(zx[0], wf_f[0], f0);  f0 = wmma4(zx[1], wf_f[1], f0);
    f1 = wmma4(zx[2], wf_f[2], f1);  f1 = wmma4(zx[3], wf_f[3], f1);
    fb0 = wmma4(z8, wfb_f[8], fb0);  fb1 = wmma4(z9, wfb_f[9], fb1);
    gb0 = wmma4(z8, wgb_f[8], gb0);  gb1 = wmma4(z9, wgb_f[9], gb1);
    f0  = wmma4(z8, wf_f[4],  f0);   f1  = wmma4(z9, wf_f[5],  f1);

    // ---- elementwise update (gamma-switch logic) ----
    v8f xfu, xbu;
#pragma unroll
    for (int r = 0; r < 8; ++r) {
      float gb  = 0.1f * fast_tanh(gb0[r] + gb1[r]);
      float gp  = sg[r];
      float igp = igpv[r];
      float dev = devv[r];
      float dq  = gb / gp;                       // exact; overlaps logic below
      float delta = (gb * (dev + 0.5f * gb * (1.0f + igp))) * (1.0f - igp);
      float eps = e1v[r]
                + dev * (dev + gb * (1.0f + igp)
                         + 0.5f * gb * gb * (1.0f + igp * igp));
      bool D1 = (eps >= delta) && (delta >= 0.0f);
      bool D2 = (eps <  delta) && (delta <  0.0f);
      bool D3 = (delta < eps)  && (eps   <  0.0f);
      float q = delta * delta - eps * eps;
      bool D4 = (d4c * eps >= q) && (q > 0.0f);
      bool sel = D1 | D2 | D3 | D4;              // gamma = sel ? gp : 1
      xbu[r] = xb[r] + (sel ? dq : gb);
      xfu[r] = xf[r] + 0.1f * fast_tanh(f0[r] + f1[r] + bft)
                     + 0.1f * fast_tanh(fb0[r] + fb1[r]);
    }

    // ---- stage PRE-clamp updated state; its fragments feed y AND (after
    //      elementwise clamp, which commutes with relayout) the next step ----
    __syncthreads();   // WAR vs this step's fragment loads
#pragma unroll
    for (int r = 0; r < 8; ++r) {
      int row = r + 8 * h;
      buf[0][row * LSTR + n] = xfu[r];
      buf[1][row * LSTR + n] = xbu[r];
    }
    __syncthreads();
#pragma unroll
    for (int i = 0; i < 4; ++i) {
      zx[i] = *reinterpret_cast<const v2f*>(&buf[0][m * LSTR + 4 * i + 2 * h]);
      zb[i] = *reinterpret_cast<const v2f*>(&buf[1][m * LSTR + 4 * i + 2 * h]);
    }
    v8f y0a = {}, y1a = {};
#pragma unroll
    for (int i = 0; i < 4; ++i) {
      y0a = wmma4(zx[i], wc_f[i],     y0a);
      y1a = wmma4(zb[i], wc_f[4 + i], y1a);
    }

    // next iteration's A-fragments = clamp(pre-clamp fragments)
#pragma unroll
    for (int i = 0; i < 4; ++i) {
      zx[i].x = clampf(zx[i].x, cminx,  cmaxx);
      zx[i].y = clampf(zx[i].y, cminx,  cmaxx);
      zb[i].x = clampf(zb[i].x, cminxb, cmaxxb);
      zb[i].y = clampf(zb[i].y, cminxb, cmaxxb);
    }

    // ---- clamp register state + running mean/std (uses clamped xb) ----
#pragma unroll
    for (int r = 0; r < 8; ++r) {
      xf[r] = clampf(xfu[r], cminx,  cmaxx);
      xb[r] = clampf(xbu[r], cminxb, cmaxxb);
      y[r]  = clampf(y0a[r] + y1a[r] + th, cminy, cmaxy);
      float devn = xb[r] - mu[r];
      float s2   = c1 * sg[r] * sg[r] + c2 * devn * devn;
      float mun  = c1 * mu[r] + inv2 * xb[r];
      sg[r] = sqrtf(s2);
      mu[r] = mun;
    }

    pxf += 16; pxb += 16; py += 8; pu += 8;
  }
}

extern "C" void kernel_launch(void* const* d_in, const int* in_sizes, int n_in,
                              void* d_out, int out_size, void* d_ws, size_t ws_size,
                              hipStream_t stream) {
  (void)in_sizes; (void)n_in; (void)out_size; (void)d_ws; (void)ws_size;
  ssm_scan_kernel<<<dim3(B_ / 16), dim3(32), 0, stream>>>(
      (const float*)d_in[0],   // x_f_0
      (const float*)d_in[1],   // x_b_0
      (const float*)d_in[2],   // u
      (const float*)d_in[3],   // theta_f
      (const float*)d_in[4],   // theta_h
      (const float*)d_in[5],   // datascaler
      (const float*)d_in[6],   // Wf
      (const float*)d_in[7],   // bf
      (const float*)d_in[8],   // Wfb
      (const float*)d_in[9],   // Wgb
      (const float*)d_in[10],  // Wh
      (const float*)d_in[11],  // Whb
      (const float*)d_in[12],  // min_cons_hard_x_b_step
      (const float*)d_in[13],  // max_cons_hard_x_b_step
      (const float*)d_in[14],  // min_cons_hard_x_step
      (const float*)d_in[15],  // max_cons_hard_x_step
      (const float*)d_in[16],  // min_cons_hard_y_step
      (const float*)d_in[17],  // max_cons_hard_y_step
      (float*)d_out);
}